// ToRMS_41077067219038
// MI455X (gfx1250) — compile-verified
//
#include <hip/hip_runtime.h>
#include <hip/hip_bf16.h>
#include <math.h>

// ---- problem constants (from reference) ----
#define FRAME_LEN 2048
#define HOP       512
#define N_SAMP    960000
#define N_ROWS    64
#define N_BLOCKS  1875        // real 512-sample block sums per row
#define N_FRAMES  1876        // output frames per row
#define GROUPS_PER_ROW 118    // ceil(1875 / 16)
#define WS_PITCH  1888        // 118*16, row pitch of block-sum workspace

#define WAVES_PER_WG 8
#define LDS_PITCH 34          // floats per block-row in LDS (32 lanes + pad, conflict-free)

typedef float v8f  __attribute__((ext_vector_type(8)));
typedef float v2f  __attribute__((ext_vector_type(2)));
typedef float vf4  __attribute__((ext_vector_type(4)));

// -------- Pass 1: per-block sum of squares, WMMA cross-lane reduction --------
__global__ __launch_bounds__(256) void rms_block_sums(const float* __restrict__ y,
                                                      float* __restrict__ S) {
    __shared__ __align__(16) float P[WAVES_PER_WG][16 * LDS_PITCH];

    const int lane     = threadIdx.x & 31;
    const int waveInWG = threadIdx.x >> 5;
    const int gwave    = blockIdx.x * WAVES_PER_WG + waveInWG;
    const int row      = gwave / GROUPS_PER_ROW;
    const int grp      = gwave % GROUPS_PER_ROW;
    const int block0   = grp * 16;

    const float* __restrict__ yr = y + (long long)row * N_SAMP;
    float* lds = P[waveInWG];

    // Each wave: 16 blocks; per block, 32 lanes x 4 float4 = 512 floats,
    // fully coalesced 512B-per-instruction streaming loads (non-temporal).
    for (int b = 0; b < 16; ++b) {
        const int blk = block0 + b;             // wave-uniform
        float partial = 0.f;
        if (blk < N_BLOCKS) {
            const vf4* src = (const vf4*)(yr + blk * HOP);
#pragma unroll
            for (int i = 0; i < 4; ++i) {
                vf4 v = __builtin_nontemporal_load(&src[i * 32 + lane]);
                partial = fmaf(v.x, v.x, partial);
                partial = fmaf(v.y, v.y, partial);
                partial = fmaf(v.z, v.z, partial);
                partial = fmaf(v.w, v.w, partial);
            }
        }
        lds[b * LDS_PITCH + lane] = partial;    // P[block][lane]
    }
    __syncthreads();

    // Cross-lane reduction with V_WMMA_F32_16X16X4_F32, B = ones (layout-immune):
    //   WMMA j covers source lanes 4j..4j+3 (the K dimension).
    //   A layout (ISA 7.12.2): lanes 0-15 hold (M=lane, K=0..1) in VGPR0/1,
    //   lanes 16-31 hold (M=lane-16, K=2..3).
    const int m    = lane & 15;
    const int koff = (lane < 16) ? 0 : 2;
    v8f c = {};
    const v2f ones = {1.f, 1.f};
#pragma unroll
    for (int j = 0; j < 8; ++j) {
        v2f a = *(const v2f*)(lds + m * LDS_PITCH + 4 * j + koff); // 8B-aligned ds_load_b64
        c = __builtin_amdgcn_wmma_f32_16x16x4_f32(false, a, false, ones,
                                                  (short)0, c, false, false);
    }

    // D layout: VGPR v = block v on lanes 0-15, block 8+v on lanes 16-31
    // (replicated over N); extract from one lane of each half.
    if (lane == 0 || lane == 16) {
        const int bbase = block0 + ((lane == 16) ? 8 : 0);
#pragma unroll
        for (int i = 0; i < 8; ++i) {
            const int blk = bbase + i;
            if (blk < N_BLOCKS) S[row * WS_PITCH + blk] = c[i];
        }
    }
}

// -------- Pass 2: window = 4 adjacent block sums (zeros outside), sqrt --------
__global__ __launch_bounds__(256) void rms_finalize(const float* __restrict__ S,
                                                    float* __restrict__ out) {
    const int idx = blockIdx.x * blockDim.x + threadIdx.x;
    if (idx >= N_ROWS * N_FRAMES) return;
    const int row = idx / N_FRAMES;
    const int j   = idx % N_FRAMES;
    const float* Sr = S + row * WS_PITCH;
    float s = 0.f;
#pragma unroll
    for (int d = 0; d < 4; ++d) {
        const int t = j - 2 + d;               // padded blocks j..j+3 -> real j-2..j+1
        if (t >= 0 && t < N_BLOCKS) s += Sr[t];
    }
    out[idx] = sqrtf(s * (1.0f / (float)FRAME_LEN));
}

extern "C" void kernel_launch(void* const* d_in, const int* in_sizes, int n_in,
                              void* d_out, int out_size, void* d_ws, size_t ws_size,
                              hipStream_t stream) {
    const float* y  = (const float*)d_in[0];
    float*       out = (float*)d_out;
    float*       S   = (float*)d_ws;            // 64 * 1888 * 4 B = 483,328 B

    const int waves = N_ROWS * GROUPS_PER_ROW;  // 7552
    const int wgs   = waves / WAVES_PER_WG;     // 944 workgroups of 256 threads
    rms_block_sums<<<wgs, 256, 0, stream>>>(y, S);

    const int n2 = N_ROWS * N_FRAMES;           // 120064
    rms_finalize<<<(n2 + 255) / 256, 256, 0, stream>>>(S, out);
}